// Kanvolution2d_44581760532960
// MI455X (gfx1250) — compile-verified
//
#include <hip/hip_runtime.h>

typedef __attribute__((ext_vector_type(2))) float v2f;
typedef __attribute__((ext_vector_type(8))) float v8f;

#define KDIM 144   // C * 3 * 3
#define C_   16
#define HW   32
#define L_   1024  // 32*32 output pixels
#define O_   32

// One wave computes a 16(o) x 16(l) tile for one batch b.
// Per k (144 of them): two V_WMMA_F32_16X16X4_F32 evaluate the degree-3
// numerator (pre-scaled by w_outer) and the denominator polynomial for the
// whole 16x16 tile; VALU+trans pipe then accumulate P'/(1+|Q|) via
// v_rcp_f32 + one Newton-Raphson step (denominator >= 1, so rcp is safe).
__global__ __launch_bounds__(128) void kanv2d_wmma_kernel(
    const float* __restrict__ x,     // [B, C, 32, 32]
    const float* __restrict__ wp,    // [O, 4, KDIM]
    const float* __restrict__ wq,    // [O, 2, KDIM]
    const float* __restrict__ wout,  // [O, KDIM]
    float* __restrict__ out)         // [B, O, 1024]
{
    const int lane  = threadIdx.x & 31;
    const int wave  = blockIdx.x * 4 + (threadIdx.x >> 5);

    // wave -> (b, o-tile, l-tile)
    const int ltile = wave & 63;          // 64 l-tiles of 16
    const int otile = (wave >> 6) & 1;    // 2 o-tiles of 16
    const int b     = wave >> 7;          // 4 batches

    const int l_base  = ltile * 16;
    const int oh      = l_base >> 5;      // 32 px per row; 16-tile never crosses a row
    const int ow_base = l_base & 31;      // 0 or 16
    const int n       = lane & 15;        // pixel col within tile AND o row within tile
    const bool hi     = lane >= 16;       // K-half selector for A/B layouts

    // Per-lane A-row pointers: lane (and lane+16) both serve matrix row M = n.
    const int   o  = otile * 16 + n;
    const float* pA = wp   + o * (4 * KDIM);
    const float* qA = wq   + o * (2 * KDIM);
    const float* oW = wout + o * KDIM;

    v8f acc = {};

    for (int c = 0; c < C_; ++c) {
        const float* xc = x + ((b * C_ + c) * HW) * HW;
        #pragma unroll
        for (int t = 0; t < 9; ++t) {
            const int k  = c * 9 + t;
            const int ki = t / 3;
            const int kj = t % 3;

            // ---------- B matrix: Vandermonde [1, x, x^2, x^3] ----------
            // unfold with pad=1: x_in = x[c, oh+ki-1, ow+kj-1], 0 if OOB.
            // Load with clamped index * valid-mask so EXEC stays all-ones.
            int ih = oh + ki - 1;
            int iw = ow_base + n + kj - 1;
            bool valid = ((unsigned)ih < 32u) && ((unsigned)iw < 32u);
            int ihc = ih < 0 ? 0 : (ih > 31 ? 31 : ih);
            int iwc = iw < 0 ? 0 : (iw > 31 ? 31 : iw);
            float xv = xc[ihc * HW + iwc] * (valid ? 1.0f : 0.0f);
            float x2 = xv * xv;
            float x3 = x2 * xv;
            v2f bm;                       // lanes 0-15: rows K=0,1 ; lanes 16-31: rows K=2,3
            bm.x = hi ? x2 : 1.0f;
            bm.y = hi ? x3 : xv;

            // ---------- A matrix (numerator), pre-scaled by w_outer ----------
            float wov = oW[k];
            float ap0 = pA[k];
            float ap1 = pA[k + KDIM];
            float ap2 = pA[k + 2 * KDIM];
            float ap3 = pA[k + 3 * KDIM];
            v2f am;                       // lanes 0-15: K=0,1 ; lanes 16-31: K=2,3
            am.x = (hi ? ap2 : ap0) * wov;
            am.y = (hi ? ap3 : ap1) * wov;

            // ---------- A matrix (denominator): coeffs [0, wq0, wq1, 0] ----------
            float aq0 = qA[k];            // coeff of x^1
            float aq1 = qA[k + KDIM];     // coeff of x^2
            v2f amq;
            amq.x = hi ? aq1 : 0.0f;      // K=0 -> 0   | K=2 -> wq1
            amq.y = hi ? 0.0f : aq0;      // K=1 -> wq0 | K=3 -> 0

            v8f zero = {};
            // D = A(16x4) * B(4x16) + 0 : evaluates the polynomial for the
            // full 16(o) x 16(l) tile in one matrix op.
            v8f P = __builtin_amdgcn_wmma_f32_16x16x4_f32(
                        false, am,  false, bm, (short)0, zero, false, false);
            v8f Q = __builtin_amdgcn_wmma_f32_16x16x4_f32(
                        false, amq, false, bm, (short)0, zero, false, false);

            #pragma unroll
            for (int r = 0; r < 8; ++r) {
                // d = 1 + |Q| >= 1: fast reciprocal on the trans pipe plus one
                // Newton-Raphson step (~0.5 ulp) instead of the IEEE div chain.
                float d  = 1.0f + __builtin_fabsf(Q[r]);
                float rc = __builtin_amdgcn_rcpf(d);
                float e  = __builtin_fmaf(-d, rc, 1.0f);
                rc       = __builtin_fmaf(e, rc, rc);
                acc[r]   = __builtin_fmaf(P[r], rc, acc[r]);
            }
        }
    }

    // D layout: reg r, lanes 0-15 -> row r ; lanes 16-31 -> row 8+r.
    const int row_base = otile * 16 + (hi ? 8 : 0);
    float* outp = out + (b * O_ + row_base) * L_ + l_base + n;
    #pragma unroll
    for (int r = 0; r < 8; ++r) {
        outp[r * L_] = acc[r];
    }
}

extern "C" void kernel_launch(void* const* d_in, const int* in_sizes, int n_in,
                              void* d_out, int out_size, void* d_ws, size_t ws_size,
                              hipStream_t stream) {
    (void)in_sizes; (void)n_in; (void)d_ws; (void)ws_size; (void)out_size;
    const float* x    = (const float*)d_in[0];
    const float* wp   = (const float*)d_in[1];
    const float* wq   = (const float*)d_in[2];
    const float* wout = (const float*)d_in[3];
    float* out = (float*)d_out;

    // 512 waves total = 4(b) * 2(o-tiles) * 64(l-tiles); 4 waves per block.
    kanv2d_wmma_kernel<<<128, 128, 0, stream>>>(x, wp, wq, wout, out);
}